// RNNAssembly_33019708572298
// MI455X (gfx1250) — compile-verified
//
#include <hip/hip_runtime.h>

typedef __attribute__((ext_vector_type(2))) float v2f;
typedef __attribute__((ext_vector_type(8))) float v8f;

#define WMMA_F32(a, b, c) \
  __builtin_amdgcn_wmma_f32_16x16x4_f32(false, (a), false, (b), (short)0, (c), false, false)

// CDNA5 async global->LDS copy (ASYNCcnt-tracked, bypasses VGPRs).
// lds_off = byte offset into LDS (low 32 bits of generic shared pointer).
__device__ __forceinline__ void async_load_b128(const void* lds_ptr, const void* gaddr) {
  asm volatile("global_load_async_to_lds_b128 %0, %1, off"
               :
               : "v"((unsigned)(unsigned long long)lds_ptr),
                 "v"((unsigned long long)gaddr)
               : "memory");
}
__device__ __forceinline__ void wait_async0() {
  asm volatile("s_wait_asynccnt 0" ::: "memory");
}

// ---------------------------------------------------------------------------
// Generic fp32 WMMA GEMM: C[M,N] = A[M,K] * B[K,N], all row-major.
// Requires M%64==0, N%64==0, K%32==0. 256 threads = 8 waves; each wave owns
// a 16x32 slab of the 64x64 block tile. A tile staged via async-to-LDS; B tile
// staged transposed (N-major) so every B fragment is one ds_load_b64.
// ---------------------------------------------------------------------------
__global__ __launch_bounds__(256) void wmma_gemm_f32(
    const float* __restrict__ A, const float* __restrict__ B, float* __restrict__ C,
    int M, int N, int K) {
  constexpr int LDA  = 36;  // A: 64 rows x (32+pad), conflict-free b64 reads
  constexpr int LDBT = 34;  // B^T: 64 rows (N) x (32+pad K), conflict-free b64 reads
  __shared__ float As[64 * LDA];
  __shared__ float BsT[64 * LDBT];

  const int tid  = threadIdx.x;
  const int lane = tid & 31, wave = tid >> 5;
  const int half = lane >> 4, l16 = lane & 15;
  const int bm = blockIdx.y * 64, bn = blockIdx.x * 64;
  const int m0 = 16 * (wave & 3);
  const int n0 = 32 * (wave >> 2);

  v8f acc0 = {}, acc1 = {};

  const int ar = tid >> 2, ac = (tid & 3) << 3;  // A tile: 64 rows x 32 cols
  const int br = tid >> 3, bc = (tid & 7) << 3;  // B tile: 32 rows x 64 cols

  for (int k0 = 0; k0 < K; k0 += 32) {
    // A tile: async DMA global->LDS (2 x b128 per thread)
    const float* ga = A + (size_t)(bm + ar) * K + k0 + ac;
    async_load_b128(&As[ar * LDA + ac], ga);
    async_load_b128(&As[ar * LDA + ac + 4], ga + 4);

    // B tile: load row-major, store transposed into LDS
    const float4* bg = (const float4*)(B + (size_t)(k0 + br) * N + bn + bc);
    float4 b0 = bg[0], b1 = bg[1];
    BsT[(bc + 0) * LDBT + br] = b0.x;
    BsT[(bc + 1) * LDBT + br] = b0.y;
    BsT[(bc + 2) * LDBT + br] = b0.z;
    BsT[(bc + 3) * LDBT + br] = b0.w;
    BsT[(bc + 4) * LDBT + br] = b1.x;
    BsT[(bc + 5) * LDBT + br] = b1.y;
    BsT[(bc + 6) * LDBT + br] = b1.z;
    BsT[(bc + 7) * LDBT + br] = b1.w;

    wait_async0();
    __syncthreads();

#pragma unroll
    for (int kk = 0; kk < 32; kk += 4) {
      const int kof = kk + 2 * half;  // lanes 16..31 take K+2, K+3
      v2f a   = *(const v2f*)&As[(m0 + l16) * LDA + kof];
      v2f bb0 = *(const v2f*)&BsT[(n0 + l16) * LDBT + kof];
      v2f bb1 = *(const v2f*)&BsT[(n0 + 16 + l16) * LDBT + kof];
      acc0 = WMMA_F32(a, bb0, acc0);
      acc1 = WMMA_F32(a, bb1, acc1);
    }
    __syncthreads();
  }

#pragma unroll
  for (int r = 0; r < 8; ++r) {
    const int row = bm + m0 + r + 8 * half;  // C/D: VGPR r -> M = r (+8 for hi half)
    C[(size_t)row * N + bn + n0 + l16]      = acc0[r];
    C[(size_t)row * N + bn + n0 + 16 + l16] = acc1[r];
  }
}

// ---------------------------------------------------------------------------
// Persistent recurrence kernel: 16 workgroups (one per ring block), weights
// resident in LDS (N-major so B frags are single b64 loads) for all 512 steps,
// device-scope atomic split barrier between steps.
//   drift_n = -s_n + tanh(s_n) @ B_n + s_{n-1} @ C_{n-1} - s_{n+1} @ C_n^T + u_t
//   s'_n    = s_n + 0.01 * drift_n
// The workgroup's own block state stays resident in LDS (Scur) across steps;
// only the two neighbor blocks are re-staged per step, via async-to-LDS.
// states[] is pre-filled with proj (u_t) and overwritten in place per step.
// ---------------------------------------------------------------------------
#define T_STEPS 512
#define BATCH   64
#define NBLK    16
#define BSZ     128
#define HID     2048
#define LDT     132   // 128 + pad

__global__ __launch_bounds__(256) void rnn_recurrence(
    const float* __restrict__ blocks, const float* __restrict__ coupling,
    float* __restrict__ states, unsigned int* __restrict__ ctr) {
  __shared__ float WbT[BSZ * LDT];   // B_n^T          (WbT[j][i]  =  B_n[i][j])
  __shared__ float WcpT[BSZ * LDT];  // C_{n-1}^T      (WcpT[j][i] =  C_{n-1}[i][j])
  __shared__ float WcnN[BSZ * LDT];  // -C_n row-major (WcnN[j][k] = -C_n[j][k])
  __shared__ float Scur[BATCH * LDT];
  __shared__ float Sprev[BATCH * LDT];
  __shared__ float Snext[BATCH * LDT];

  const int n    = blockIdx.x;
  const int tid  = threadIdx.x;
  const int lane = tid & 31, wave = tid >> 5;
  const int half = lane >> 4, l16 = lane & 15;
  const int m0    = 16 * (wave & 3);
  const int nbase = 64 * (wave >> 2);
  const int np = (n + NBLK - 1) % NBLK;
  const int nn = (n + 1) % NBLK;

  // ---- load weights into LDS once (resident for all 512 steps) ----
  {
    const float* Bn = blocks   + (size_t)n  * BSZ * BSZ;
    const float* Cp = coupling + (size_t)np * BSZ * BSZ;
    const float* Cn = coupling + (size_t)n  * BSZ * BSZ;
    for (int i = tid; i < BSZ * BSZ / 4; i += 256) {
      const int r = i >> 5, c = (i & 31) << 2;
      float4 vb = *(const float4*)(Bn + (size_t)r * BSZ + c);
      float4 vp = *(const float4*)(Cp + (size_t)r * BSZ + c);
      float4 vn = *(const float4*)(Cn + (size_t)r * BSZ + c);
      WbT[(c + 0) * LDT + r] = vb.x;   // transpose
      WbT[(c + 1) * LDT + r] = vb.y;
      WbT[(c + 2) * LDT + r] = vb.z;
      WbT[(c + 3) * LDT + r] = vb.w;
      WcpT[(c + 0) * LDT + r] = vp.x;  // transpose
      WcpT[(c + 1) * LDT + r] = vp.y;
      WcpT[(c + 2) * LDT + r] = vp.z;
      WcpT[(c + 3) * LDT + r] = vp.w;
      float4 vm = {-vn.x, -vn.y, -vn.z, -vn.w};  // -C_n, already N-major
      *(float4*)&WcnN[r * LDT + c] = vm;
    }
    // zero-init state staging (t==0 state is all zeros; Scur stays LDS-resident)
    for (int i = tid; i < BATCH * LDT; i += 256) {
      Scur[i]  = 0.f;
      Sprev[i] = 0.f;
      Snext[i] = 0.f;
    }
  }
  __syncthreads();

  const size_t tsz = (size_t)BATCH * HID;
  for (int t = 0; t < T_STEPS; ++t) {
    // ---- stage neighbor blocks n-1, n+1 of states[t-1] (async DMA) ----
    if (t > 0) {
      const float* sp = states + (size_t)(t - 1) * tsz;
      for (int i = tid; i < BATCH * BSZ / 4; i += 256) {
        const int r = i >> 5, c = (i & 31) << 2;
        async_load_b128(&Sprev[r * LDT + c], sp + (size_t)r * HID + np * BSZ + c);
        async_load_b128(&Snext[r * LDT + c], sp + (size_t)r * HID + nn * BSZ + c);
      }
      wait_async0();
    }
    __syncthreads();

    // ---- 3 chained GEMMs per 16x16 tile, single accumulator ----
    v8f acc[4] = {};
#pragma unroll 4
    for (int kk = 0; kk < BSZ; kk += 4) {
      const int kof = kk + 2 * half;
      v2f raw = *(const v2f*)&Scur[(m0 + l16) * LDT + kof];
      v2f aT;
      aT.x = tanhf(raw.x);
      aT.y = tanhf(raw.y);
      v2f aP = *(const v2f*)&Sprev[(m0 + l16) * LDT + kof];
      v2f aN = *(const v2f*)&Snext[(m0 + l16) * LDT + kof];
#pragma unroll
      for (int j = 0; j < 4; ++j) {
        const int ncol = nbase + 16 * j + l16;
        v2f b;
        b = *(const v2f*)&WbT[ncol * LDT + kof];
        acc[j] = WMMA_F32(aT, b, acc[j]);          // + tanh(s_n) @ B_n
        b = *(const v2f*)&WcpT[ncol * LDT + kof];
        acc[j] = WMMA_F32(aP, b, acc[j]);          // + s_{n-1} @ C_{n-1}
        b = *(const v2f*)&WcnN[ncol * LDT + kof];
        acc[j] = WMMA_F32(aN, b, acc[j]);          // - s_{n+1} @ C_n^T
      }
    }
    __syncthreads();  // all reads of Scur done before epilogue rewrites it

    // ---- epilogue: s' = s + dt*(acc - s + u); u lives in states[t] in place.
    //      New state goes to global (output) AND stays resident in Scur.  ----
    float* so = states + (size_t)t * tsz + (size_t)n * BSZ;
#pragma unroll
    for (int j = 0; j < 4; ++j) {
#pragma unroll
      for (int r = 0; r < 8; ++r) {
        const int row = m0 + r + 8 * half;
        const int col = nbase + 16 * j + l16;
        const float s = Scur[row * LDT + col];
        const float u = so[(size_t)row * HID + col];
        const float nv = s + 0.01f * (acc[j][r] - s + u);
        so[(size_t)row * HID + col] = nv;
        Scur[row * LDT + col] = nv;
      }
    }

    // ---- device-scope split barrier across the 16 workgroups ----
    __threadfence();
    __syncthreads();
    if (tid == 0) {
      __hip_atomic_fetch_add(&ctr[t], 1u, __ATOMIC_ACQ_REL, __HIP_MEMORY_SCOPE_AGENT);
      while (__hip_atomic_load(&ctr[t], __ATOMIC_ACQUIRE, __HIP_MEMORY_SCOPE_AGENT) <
             (unsigned)NBLK) {
        __builtin_amdgcn_s_sleep(2);
      }
    }
    __syncthreads();
    __threadfence();
  }
}

// ---------------------------------------------------------------------------
// in: input[512,64,128], blocks[16,128,128], coupling[16,128,128],
//     input_mat[128,2048], out_mat[2048,128]
// out: output[512,64,128] ++ states[512,64,2048]  (fp32, flat)
// ---------------------------------------------------------------------------
extern "C" void kernel_launch(void* const* d_in, const int* in_sizes, int n_in,
                              void* d_out, int out_size, void* d_ws, size_t ws_size,
                              hipStream_t stream) {
  const float* input     = (const float*)d_in[0];
  const float* blocks    = (const float*)d_in[1];
  const float* coupling  = (const float*)d_in[2];
  const float* input_mat = (const float*)d_in[3];
  const float* out_mat   = (const float*)d_in[4];

  float* out    = (float*)d_out;
  float* states = out + (size_t)T_STEPS * BATCH * 128;  // states region of d_out
  unsigned int* ctr = (unsigned int*)d_ws;

  const int M = T_STEPS * BATCH;  // 32768

  // barrier counters must start at zero every call (graph-capturable memset)
  hipMemsetAsync(ctr, 0, T_STEPS * sizeof(unsigned int), stream);

  // 1) proj = input @ input_mat -> states region (u_t, consumed in place)
  wmma_gemm_f32<<<dim3(HID / 64, M / 64), dim3(256), 0, stream>>>(
      input, input_mat, states, M, HID, 128);

  // 2) persistent recurrence over 512 steps
  rnn_recurrence<<<dim3(NBLK), dim3(256), 0, stream>>>(blocks, coupling, states, ctr);

  // 3) output = states @ out_mat
  wmma_gemm_f32<<<dim3(128 / 64, M / 64), dim3(256), 0, stream>>>(
      states, out_mat, out, M, 128, HID);
}